// CR8_reg_cond_mul_13975823582038
// MI455X (gfx1250) — compile-verified
//
#include <hip/hip_runtime.h>
#include <math.h>
#include <stdint.h>

typedef float v2f __attribute__((ext_vector_type(2)));
typedef float v8f __attribute__((ext_vector_type(8)));
typedef unsigned int u32x4 __attribute__((ext_vector_type(4)));
typedef int i32x4 __attribute__((ext_vector_type(4)));
typedef int i32x8 __attribute__((ext_vector_type(8)));

#define WPIX     131072
#define CH       128
#define TILE_P   128
#define TILES_PER_WG 8
#define NTHREADS 256
#define LSTRIDE  132   // LDS row stride in floats (TDM pad: 128 dw + 4 dw pad)

// LDS layout (float offsets)
#define OFF_W1 0                          // 128 x 132 : conv1 weights
#define OFF_W2 (OFF_W1 + 128 * LSTRIDE)   // 144 x 132 : conv2 weights, rows 129..143 zero
#define OFF_XS (OFF_W2 + 144 * LSTRIDE)   // 128 x 132 : x_latent, c-major (B operand of GEMM2)
#define OFF_SC (OFF_XS + 128 * LSTRIDE)   // 128 : folded BN scale
#define OFF_SH (OFF_SC + 128)             // 128 : folded BN shift (incl conv1 bias)
#define OFF_B2 (OFF_SH + 128)             // 132 : conv2 bias (129 used, rest 0)
#define LDS_FLOATS (OFF_B2 + 132)

// ---- Tensor Data Mover: 2D weight tile (rows x 128 f32) -> LDS with 132-float row stride ----
__device__ __forceinline__ void tdm_load_weights(const float* gsrc, unsigned lds_byte, int rows) {
  unsigned long long ga = (unsigned long long)(uintptr_t)gsrc;
  u32x4 g0;
  g0.x = 1u;                                                // count=1 (valid user descriptor)
  g0.y = lds_byte;                                          // D#.lds_addr
  g0.z = (unsigned)(ga & 0xFFFFFFFFu);                      // global_addr[31:0]
  g0.w = (unsigned)((ga >> 32) & 0x1FFFFFFu) | (2u << 30);  // global_addr[56:32] | type=2
  i32x8 g1;
  // data_size=2 (4B) | pad_enable | pad_interval=6 (128 DW) | pad_amount=3 (4 DW)
  g1[0] = (2 << 16) | (1 << 20) | (6 << 22) | (3 << 25);
  g1[1] = (CH & 0xFFFF) << 16;          // tensor_dim0[15:0]  (bits 63:48)
  g1[2] = (rows & 0xFFFF) << 16;        // tensor_dim0 hi = 0; tensor_dim1[15:0] (bits 95:80)
  g1[3] = (CH & 0xFFFF) << 16;          // tensor_dim1 hi = 0; tile_dim0 = 128 (bits 127:112)
  g1[4] = (rows & 0xFFFF);              // tile_dim1 = rows; tile_dim2 = 0
  g1[5] = CH;                           // tensor_dim0_stride[31:0] = 128
  g1[6] = 0;                            // stride hi / tensor_dim1_stride (unused for 2D)
  g1[7] = 0;
  i32x4 z4 = {0, 0, 0, 0};
#if __clang_major__ >= 23
  i32x8 z8 = {0, 0, 0, 0, 0, 0, 0, 0};
  __builtin_amdgcn_tensor_load_to_lds(g0, g1, z4, z4, z8, 0);
#else
  __builtin_amdgcn_tensor_load_to_lds(g0, g1, z4, z4, 0);
#endif
}

__launch_bounds__(NTHREADS, 1)
__global__ void CR8_fused_head_kernel(
    const float* __restrict__ x,
    const float* __restrict__ w1,
    const float* __restrict__ b1c,
    const float* __restrict__ gamma,
    const float* __restrict__ beta,
    const float* __restrict__ mean,
    const float* __restrict__ var,
    const float* __restrict__ w2,
    const float* __restrict__ b2c,
    const float* __restrict__ cmw,
    const float* __restrict__ cmb,
    float* __restrict__ out0,
    float* __restrict__ outm)
{
  __shared__ __align__(16) float lds[LDS_FLOATS];
  const int t = threadIdx.x;

  // ---- One-shot staging per workgroup (amortized over 8 pixel tiles) ----
  if (t == 0) {
    // TDM DMA: W1 (128x128) and W2 (129x128) -> LDS, 132-float row stride via pad
    tdm_load_weights(w1, (unsigned)(uintptr_t)&lds[OFF_W1], 128);
    tdm_load_weights(w2, (unsigned)(uintptr_t)&lds[OFF_W2], 129);
    __builtin_amdgcn_s_wait_tensorcnt(0);
  }
  // Zero-pad W2 rows 129..143 (makes mask row a clean 9th WMMA M-tile)
  {
    float4 zv = make_float4(0.f, 0.f, 0.f, 0.f);
    for (int i4 = t; i4 < (15 * 128 / 4); i4 += NTHREADS) {
      int e = i4 << 2;
      int r = 129 + (e >> 7), c = e & 127;
      *(float4*)&lds[OFF_W2 + r * LSTRIDE + c] = zv;
    }
  }
  // Fold conv1 bias + BN into per-channel scale/shift
  if (t < 128) {
    float sc = gamma[t] * rsqrtf(var[t] + 1e-5f);
    lds[OFF_SC + t] = sc;
    lds[OFF_SH + t] = (b1c[t] - mean[t]) * sc + beta[t];
  }
  if (t < 132) lds[OFF_B2 + t] = (t < 129) ? b2c[t] : 0.f;
  __syncthreads();

  const int lane = t & 31;          // wave32
  const int wv   = t >> 5;          // wave id 0..7
  const int hi   = lane >> 4;       // half-wave select (K-split / M-split)
  const int ln   = lane & 15;       // pixel (N) / row (M) within 16
  const int pw   = wv * 16;         // wave's pixel base in tile

  const int tile0 = blockIdx.x * TILES_PER_WG;

  float bxv[32], byv[32];   // X columns, B operand of GEMM1 (next-tile prefetched)
  float cxv[32], cyv[32];   // latent columns, B operand of GEMM2

  // Prologue: load X B-operand for first tile (coalesced: lanes -> consecutive w)
  {
    const int pg = tile0 * TILE_P + pw + ln;
    const float* xc = x + (size_t)(pg >> 17) * CH * WPIX + (pg & (WPIX - 1));
    #pragma unroll
    for (int k = 0; k < 32; ++k) {
      int c = 4 * k + 2 * hi;
      bxv[k] = xc[(size_t)c * WPIX];
      byv[k] = xc[(size_t)(c + 1) * WPIX];
    }
  }

  #pragma unroll 1
  for (int it = 0; it < TILES_PER_WG; ++it) {
    const int tile = tile0 + it;
    const int pg   = tile * TILE_P + pw + ln;   // this lane's pixel (current tile)

    // ---- GEMM1: h = W1 * x, fused BN + LeakyReLU -> latent in LDS (c-major) ----
    #pragma unroll 1
    for (int mt = 0; mt < 8; ++mt) {
      const float* al = &lds[OFF_W1 + (mt * 16 + ln) * LSTRIDE + 2 * hi];
      v8f acc = {0.f, 0.f, 0.f, 0.f, 0.f, 0.f, 0.f, 0.f};
      #pragma unroll
      for (int k = 0; k < 32; ++k) {
        v2f a; a.x = al[4 * k]; a.y = al[4 * k + 1];
        v2f b; b.x = bxv[k];    b.y = byv[k];
        acc = __builtin_amdgcn_wmma_f32_16x16x4_f32(false, a, false, b,
                                                    (short)0, acc, false, false);
      }
      #pragma unroll
      for (int r = 0; r < 8; ++r) {
        int o = mt * 16 + 8 * hi + r;          // C layout: VGPR r -> M = r + 8*hi
        float h  = acc[r] * lds[OFF_SC + o] + lds[OFF_SH + o];
        float xl = fmaxf(h, 0.01f * h);        // LeakyReLU(0.01)
        lds[OFF_XS + o * LSTRIDE + pw + ln] = xl;
      }
    }

    // ---- Software pipeline: issue next tile's X loads; latency hides under GEMM2 ----
    if (it + 1 < TILES_PER_WG) {
      const int png = (tile + 1) * TILE_P + pw + ln;
      const float* xc = x + (size_t)(png >> 17) * CH * WPIX + (png & (WPIX - 1));
      #pragma unroll
      for (int k = 0; k < 32; ++k) {
        int c = 4 * k + 2 * hi;
        bxv[k] = xc[(size_t)c * WPIX];
        byv[k] = xc[(size_t)(c + 1) * WPIX];
      }
    }

    // ---- B operand of GEMM2 from latent LDS (wave-private columns) ----
    #pragma unroll
    for (int k = 0; k < 32; ++k) {
      int c = 4 * k + 2 * hi;
      cxv[k] = lds[OFF_XS + c * LSTRIDE + pw + ln];
      cyv[k] = lds[OFF_XS + (c + 1) * LSTRIDE + pw + ln];
    }

    float bestv = -3.402823466e38f;
    int   besto = 0;

    // ---- GEMM2 class tiles (rows 0..127) + running argmax ----
    #pragma unroll 1
    for (int mt = 0; mt < 8; ++mt) {
      const float* al = &lds[OFF_W2 + (mt * 16 + ln) * LSTRIDE + 2 * hi];
      v8f acc = {0.f, 0.f, 0.f, 0.f, 0.f, 0.f, 0.f, 0.f};
      #pragma unroll
      for (int k = 0; k < 32; ++k) {
        v2f a; a.x = al[4 * k]; a.y = al[4 * k + 1];
        v2f b; b.x = cxv[k];    b.y = cyv[k];
        acc = __builtin_amdgcn_wmma_f32_16x16x4_f32(false, a, false, b,
                                                    (short)0, acc, false, false);
      }
      #pragma unroll
      for (int r = 0; r < 8; ++r) {
        int o = mt * 16 + 8 * hi + r;          // ascending o per lane -> first-max kept
        float v = acc[r] + lds[OFF_B2 + o];
        if (v > bestv) { bestv = v; besto = o; }
      }
    }

    // ---- GEMM2 mask tile (row 128 real, rows 129..143 zero-padded) ----
    {
      const float* al = &lds[OFF_W2 + (128 + ln) * LSTRIDE + 2 * hi];
      v8f acc = {0.f, 0.f, 0.f, 0.f, 0.f, 0.f, 0.f, 0.f};
      #pragma unroll
      for (int k = 0; k < 32; ++k) {
        v2f a; a.x = al[4 * k]; a.y = al[4 * k + 1];
        v2f b; b.x = cxv[k];    b.y = cyv[k];
        acc = __builtin_amdgcn_wmma_f32_16x16x4_f32(false, a, false, b,
                                                    (short)0, acc, false, false);
      }
      float m = acc[0] + lds[OFF_B2 + 128];    // tile-row 0 lives in VGPR0 lanes 0..15
      float msk = fmaxf(m, 0.01f * m);
      if (hi == 0) outm[pg] = msk;
    }

    // ---- Combine argmax across the two half-wave lanes of each pixel ----
    {
      float ov = __shfl_xor(bestv, 16, 32);
      int   oo = __shfl_xor(besto, 16, 32);
      if (ov > bestv || (ov == bestv && oo < besto)) { bestv = ov; besto = oo; }
    }

    // ---- Class-conditional regression: dot(x_latent[:,p], cm_w[besto]) ----
    const float4* cmrow = (const float4*)(cmw + (size_t)besto * CH) + hi * 16;
    float part = 0.f;
    #pragma unroll
    for (int i4 = 0; i4 < 16; ++i4) {
      float4 cw = cmrow[i4];
      int c = hi * 64 + (i4 << 2);
      part = fmaf(cw.x, lds[OFF_XS + (c + 0) * LSTRIDE + pw + ln], part);
      part = fmaf(cw.y, lds[OFF_XS + (c + 1) * LSTRIDE + pw + ln], part);
      part = fmaf(cw.z, lds[OFF_XS + (c + 2) * LSTRIDE + pw + ln], part);
      part = fmaf(cw.w, lds[OFF_XS + (c + 3) * LSTRIDE + pw + ln], part);
    }
    float tot = part + __shfl_xor(part, 16, 32);
    if (hi == 0) {
      float reg = tot + cmb[besto];
      out0[pg] = ((float)besto + reg) * (1.0f / 128.0f);
    }
  }
}

extern "C" void kernel_launch(void* const* d_in, const int* in_sizes, int n_in,
                              void* d_out, int out_size, void* d_ws, size_t ws_size,
                              hipStream_t stream) {
  (void)in_sizes; (void)n_in; (void)out_size; (void)d_ws; (void)ws_size;
  const float* x     = (const float*)d_in[0];
  const float* w1    = (const float*)d_in[1];
  const float* b1c   = (const float*)d_in[2];
  const float* gamma = (const float*)d_in[3];
  const float* beta  = (const float*)d_in[4];
  const float* mean  = (const float*)d_in[5];
  const float* var   = (const float*)d_in[6];
  const float* w2    = (const float*)d_in[7];
  const float* b2c   = (const float*)d_in[8];
  const float* cmw   = (const float*)d_in[9];
  const float* cmb   = (const float*)d_in[10];

  const int npix = 4 * WPIX;                         // B*H*W = 524288
  float* out0 = (float*)d_out;
  float* outm = out0 + npix;

  dim3 grid(npix / (TILE_P * TILES_PER_WG));         // 512 workgroups x 8 tiles
  dim3 block(NTHREADS);                              // 8 waves (wave32)
  CR8_fused_head_kernel<<<grid, block, 0, stream>>>(
      x, w1, b1c, gamma, beta, mean, var, w2, b2c, cmw, cmb, out0, outm);
}